// HuberEMA_27281632264624
// MI455X (gfx1250) — compile-verified
//
#include <hip/hip_runtime.h>
#include <cstdint>
#include <cstddef>

// Problem constants (match reference): x is (B,T,C) f32, out is (B,T,C) f32.
#define B_    32
#define T_    2048
#define C_    512
#define TPB   128            // threads per block == channels per block
#define TT    64             // time steps per LDS tile
#define NBUF  3              // triple buffering
#define ROUNDS 16            // (TT*TPB*4 bytes) / (TPB threads * 16 bytes)
#define NTILES (T_ / TT)     // 32

#if defined(__HIP_DEVICE_COMPILE__)
typedef int v4i __attribute__((vector_size(16)));
typedef __attribute__((address_space(1))) v4i GV4;   // global int4
typedef __attribute__((address_space(3))) v4i LV4;   // LDS int4
#endif

// 16B-per-lane async DMA: global -> LDS, tracked by ASYNCcnt (CDNA5).
__device__ __forceinline__ void async_cp16(const float* gsrc, float* ldst) {
#if defined(__HIP_DEVICE_COMPILE__)
#if __has_builtin(__builtin_amdgcn_global_load_async_to_lds_b128)
  __builtin_amdgcn_global_load_async_to_lds_b128((GV4*)gsrc, (LV4*)ldst,
                                                 /*offset=*/0, /*cpol=*/0);
#else
  unsigned loff = (unsigned)(uintptr_t)(LV4*)ldst;   // LDS byte offset
  asm volatile("global_load_async_to_lds_b128 %0, %1, off"
               :: "v"(loff), "v"(gsrc) : "memory");
#endif
#endif
}

// Wait until at most 2*ROUNDS (=32) async ops remain outstanding for this wave:
// the 16 oldest (the tile about to be consumed) have landed in LDS; the two
// prefetch tiles stay in flight.
__device__ __forceinline__ void wait_async_2tiles() {
#if defined(__HIP_DEVICE_COMPILE__)
#if __has_builtin(__builtin_amdgcn_s_wait_asynccnt)
  __builtin_amdgcn_s_wait_asynccnt(2 * ROUNDS);
#else
  asm volatile("s_wait_asynccnt 32" ::: "memory");
#endif
#endif
}

// Raw workgroup split-barrier (no workgroup fence => no compiler-inserted
// s_wait_asynccnt 0 that would drain the prefetch pipeline). The "memory"
// clobber stops the compiler from moving LDS accesses across it; data
// visibility for the async DMA writes is provided by wait_async_2tiles().
__device__ __forceinline__ void wg_barrier() {
#if defined(__HIP_DEVICE_COMPILE__)
  asm volatile("s_barrier_signal -1\n\ts_barrier_wait -1" ::: "memory");
#endif
}

__global__ __launch_bounds__(TPB, 1)
void huber_ema_scan(const float* __restrict__ x,
                    const float* __restrict__ logit_alpha,
                    const float* __restrict__ delta,
                    float* __restrict__ out) {
  __shared__ float buf[NBUF][TT * TPB];   // 3 x 32 KB = 96 KB LDS

  const int tid = threadIdx.x;
  const int gpb = C_ / TPB;               // channel groups per batch (4)
  const int b   = blockIdx.x / gpb;
  const int c0  = (blockIdx.x % gpb) * TPB;

  // Per-channel smoothing coefficient: k = 1 - clamp(sigmoid(la), 1e-4, 1-1e-4)
  float a = 1.0f / (1.0f + __expf(-logit_alpha[c0 + tid]));
  a = fminf(fmaxf(a, 1.0e-4f), 1.0f - 1.0e-4f);
  const float k = 1.0f - a;
  const float d = *delta;

  // Float index of element (b, t=0, c0).
  const size_t base = ((size_t)b * T_) * C_ + (size_t)c0;

  // Tile copy geometry: block moves TT x TPB floats; each round the 128
  // threads move 2048 B = 4 rows of 512 B. Thread -> (row, 4-float column).
  const int lrow = tid >> 5;          // 0..3
  const int lcol = (tid & 31) * 4;    // float offset within row

  auto issue_tile = [&](int bi, int t0) {
#pragma unroll
    for (int r = 0; r < ROUNDS; ++r) {
      const int row = lrow + r * 4;
      async_cp16(x + base + (size_t)(t0 + row) * C_ + (size_t)lcol,
                 &buf[bi][row * TPB + lcol]);
    }
  };

  // Prologue: fill the 3-deep pipeline (48 async ops in flight per wave).
  issue_tile(0, 0);
  issue_tile(1, TT);
  issue_tile(2, 2 * TT);

  float y = 0.0f;
  for (int tile = 0; tile < NTILES; ++tile) {
    wait_async_2tiles();    // my 16 oldest async loads (this tile) are done
    wg_barrier();           // => every wave's slice of this tile is in LDS

    const float* src = &buf[tile % NBUF][tid];
    float* dst = out + base + (size_t)tile * TT * C_ + (size_t)tid;

    int r0 = 0;
    if (tile == 0) {        // y[0] = x[0]
      y = src[0];
      dst[0] = y;
      r0 = 1;
    }
#pragma unroll 8
    for (int row = r0; row < TT; ++row) {
      const float xv = src[(size_t)row * TPB];          // ds_load_b32
      const float r_ = xv - y;
      const float g  = fminf(fmaxf(r_, -d), d);         // huber clip == clamp
      y = fmaf(k, g, y);
      dst[(size_t)row * C_] = y;                        // coalesced b32 store
    }

    wg_barrier();           // all waves done reading this buffer
    if (tile + NBUF < NTILES)
      issue_tile(tile % NBUF, (tile + NBUF) * TT);      // refill the buffer
  }
}

extern "C" void kernel_launch(void* const* d_in, const int* in_sizes, int n_in,
                              void* d_out, int out_size, void* d_ws, size_t ws_size,
                              hipStream_t stream) {
  (void)in_sizes; (void)n_in; (void)out_size; (void)d_ws; (void)ws_size;
  const float* x     = (const float*)d_in[0];
  const float* la    = (const float*)d_in[1];
  const float* delta = (const float*)d_in[2];
  float* out = (float*)d_out;

  dim3 grid(B_ * (C_ / TPB));   // 128 blocks
  dim3 block(TPB);              // 128 threads (4 wave32s)
  huber_ema_scan<<<grid, block, 0, stream>>>(x, la, delta, out);
}